// ProjectLayer_50843822850743
// MI455X (gfx1250) — compile-verified
//
#include <hip/hip_runtime.h>

typedef float v2f __attribute__((ext_vector_type(2)));
typedef float v8f __attribute__((ext_vector_type(8)));

#define NV 5
#define NB 4
#define NJ 15
#define HD 128
#define WD 240
#define HW (HD * WD)
#define NPTS 128000          // 80*80*20
#define CUBES_ELEMS (NB * NJ * NPTS)   // 7,680,000

__device__ __forceinline__ void grid_coord(int p, float& gx, float& gy, float& gz) {
    int ix = p / 1600;
    int rem = p - ix * 1600;
    int iy = rem / 20;
    int iz = rem - iy * 20;
    gx = -4000.0f + 101.26582278481013f * (float)ix;          // linspace + center 0
    gy = -4500.0f + 101.26582278481013f * (float)iy;          // linspace + center -500
    gz = -200.0f  + 105.26315789473684f * (float)iz;          // linspace + center 800
}

// 1-ulp hardware reciprocal (v_rcp_f32) instead of IEEE div sequence
__device__ __forceinline__ float fast_rcp(float x) {
    return __builtin_amdgcn_rcpf(x);
}

__global__ __launch_bounds__(256) void project_layer_kernel(
    const float* __restrict__ hm,     // [5,4,15,128,240]
    const float* __restrict__ camR,   // [5,4,3,3]
    const float* __restrict__ camT,   // [5,4,3]
    const float* __restrict__ camF,   // [5,4,2]
    const float* __restrict__ camC,   // [5,4,2]
    const float* __restrict__ camK,   // [5,4,3]
    const float* __restrict__ camP,   // [5,4,2]
    const float* __restrict__ ctr,    // [5,4,2]
    const float* __restrict__ scl,    // [5,4,2]
    float* __restrict__ out)          // cubes (7,680,000) ++ grids (1,536,000)
{
    const int b    = blockIdx.y;
    const int n    = blockIdx.x * 256 + threadIdx.x;     // this lane's voxel
    const int lane = threadIdx.x & 31;
    const bool hi  = lane >= 16;
    const int m    = lane & 15;
    // A-matrix row mapped to rotation row (rows 0-2 and 8-10 hold [R|t])
    const int r = (m < 3) ? m : ((m >= 8 && m < 11) ? (m - 8) : -1);

    // Own grid point
    float gx, gy, gz;
    grid_coord(n, gx, gy, gz);

    // ---- Build B matrices (view-invariant): 4x16 f32, K=(x,y,z,1), N=points ----
    // Layout: VGPR0: lanes0-15 K=0, lanes16-31 K=2 ; VGPR1: lanes0-15 K=1, lanes16-31 K=3
    const int pBase = blockIdx.x * 256 + (threadIdx.x & ~31);
    const int colN  = lane & 15;
    float c1x, c1y, c1z, c2x, c2y, c2z;
    grid_coord(pBase + colN,      c1x, c1y, c1z);   // columns for WMMA #1 (points 0..15)
    grid_coord(pBase + 16 + colN, c2x, c2y, c2z);   // columns for WMMA #2 (points 16..31)
    v2f B1, B2;
    B1.x = hi ? c1z : c1x;  B1.y = hi ? 1.0f : c1y;
    B2.x = hi ? c2z : c2x;  B2.y = hi ? 1.0f : c2y;

    float num[NJ];
#pragma unroll
    for (int j = 0; j < NJ; ++j) num[j] = 0.0f;
    float den = 0.0f;

    for (int v = 0; v < NV; ++v) {
        const int vb = v * NB + b;                    // uniform -> scalar loads
        const float* Rp = camR + vb * 9;
        const float r00 = Rp[0], r01 = Rp[1], r02 = Rp[2];
        const float r10 = Rp[3], r11 = Rp[4], r12 = Rp[5];
        const float r20 = Rp[6], r21 = Rp[7], r22 = Rp[8];
        const float T0 = camT[vb * 3 + 0], T1 = camT[vb * 3 + 1], T2 = camT[vb * 3 + 2];
        const float t0 = -(r00 * T0 + r01 * T1 + r02 * T2);
        const float t1 = -(r10 * T0 + r11 * T1 + r12 * T2);
        const float t2 = -(r20 * T0 + r21 * T1 + r22 * T2);

        // ---- Build A: 16x4 f32, rows 0-2 & 8-10 = [R | t] ----
        // Layout: VGPR0: lanes0-15 K=0, lanes16-31 K=2 ; VGPR1: lanes0-15 K=1, lanes16-31 K=3
        const float rc0 = (r == 0) ? r00 : (r == 1) ? r10 : (r == 2) ? r20 : 0.0f;
        const float rc1 = (r == 0) ? r01 : (r == 1) ? r11 : (r == 2) ? r21 : 0.0f;
        const float rc2 = (r == 0) ? r02 : (r == 1) ? r12 : (r == 2) ? r22 : 0.0f;
        const float tt  = (r == 0) ? t0  : (r == 1) ? t1  : (r == 2) ? t2  : 0.0f;
        v2f A;
        A.x = hi ? rc2 : rc0;
        A.y = hi ? tt  : rc1;

        // ---- xcam = A x B via WMMA (EXEC is all-ones: no divergence above) ----
        v8f cz = {0.f, 0.f, 0.f, 0.f, 0.f, 0.f, 0.f, 0.f};
        v8f d1 = __builtin_amdgcn_wmma_f32_16x16x4_f32(false, A, false, B1, (short)0, cz, false, false);
        v8f d2 = __builtin_amdgcn_wmma_f32_16x16x4_f32(false, A, false, B2, (short)0, cz, false, false);
        // lanes 0-15: D1 VGPR g = row M=g (cols = points pBase+lane)
        // lanes 16-31: D2 VGPR g = row M=g+8 (cols = points pBase+16+(lane-16) = own)
        const float xcx = hi ? d2[0] : d1[0];
        const float xcy = hi ? d2[1] : d1[1];
        const float xcz = hi ? d2[2] : d1[2];

        // ---- distortion + pixel projection ----
        const float invz = fast_rcp(xcz + 1e-5f);
        const float yx = xcx * invz, yy = xcy * invz;
        const float r2 = yx * yx + yy * yy;
        const float k0 = camK[vb * 3 + 0], k1 = camK[vb * 3 + 1], k2 = camK[vb * 3 + 2];
        const float p0 = camP[vb * 2 + 0], p1 = camP[vb * 2 + 1];
        const float radial = 1.0f + k0 * r2 + k1 * r2 * r2 + k2 * r2 * r2 * r2;
        const float tan2 = p0 * yy + p1 * yx;
        const float sfac = radial + 2.0f * tan2;
        const float ydx = yx * sfac + p1 * r2;
        const float ydy = yy * sfac + p0 * r2;
        float pixx = camF[vb * 2 + 0] * ydx + camC[vb * 2 + 0];
        float pixy = camF[vb * 2 + 1] * ydy + camC[vb * 2 + 1];

        // ---- bounding + clip ----
        const float ctx = ctr[vb * 2 + 0], cty = ctr[vb * 2 + 1];
        const float whx = 2.0f * ctx, why = 2.0f * cty;
        const bool bnd = (pixx >= 0.0f) & (pixy >= 0.0f) & (pixx < whx) & (pixy < why);
        const float maxwh = fmaxf(whx, why);
        pixx = fminf(fmaxf(pixx, -1.0f), maxwh);
        pixy = fminf(fmaxf(pixy, -1.0f), maxwh);

        // ---- affine to heatmap coords + [-1.1,1.1] clip ----
        const float ihsx = fast_rcp(200.0f * scl[vb * 2 + 0]);
        const float ihsy = fast_rcp(200.0f * scl[vb * 2 + 1]);
        const float tsx = 960.0f * ihsx,  tsy = 512.0f * ihsy;
        const float tox = 960.0f * (0.5f - ctx * ihsx);
        const float toy = 512.0f * (0.5f - cty * ihsy);
        const float xyx = (pixx * tsx + tox) * 0.25f;   // img -> heatmap (240/960, 128/512)
        const float xyy = (pixy * tsy + toy) * 0.25f;
        float sgx = xyx * (2.0f / 239.0f) - 1.0f;
        float sgy = xyy * (2.0f / 127.0f) - 1.0f;
        sgx = fminf(fmaxf(sgx, -1.1f), 1.1f);
        sgy = fminf(fmaxf(sgy, -1.1f), 1.1f);
        const float pxf = (sgx + 1.0f) * 119.5f;        // 0.5*(WD-1)
        const float pyf = (sgy + 1.0f) * 63.5f;         // 0.5*(HD-1)

        den += bnd ? 1.0f : 0.0f;
        if (bnd) {
            const float fx0 = floorf(pxf), fy0 = floorf(pyf);
            const float wx = pxf - fx0, wy = pyf - fy0;
            const int x0 = (int)fx0, y0 = (int)fy0;
            const int x1 = x0 + 1,   y1 = y0 + 1;
            const float v00 = (x0 >= 0 && x0 < WD && y0 >= 0 && y0 < HD) ? 1.0f : 0.0f;
            const float v01 = (x1 >= 0 && x1 < WD && y0 >= 0 && y0 < HD) ? 1.0f : 0.0f;
            const float v10 = (x0 >= 0 && x0 < WD && y1 >= 0 && y1 < HD) ? 1.0f : 0.0f;
            const float v11 = (x1 >= 0 && x1 < WD && y1 >= 0 && y1 < HD) ? 1.0f : 0.0f;
            const float w00 = (1.0f - wx) * (1.0f - wy) * v00;
            const float w01 = wx * (1.0f - wy) * v01;
            const float w10 = (1.0f - wx) * wy * v10;
            const float w11 = wx * wy * v11;
            const int cx0 = min(max(x0, 0), WD - 1), cx1 = min(max(x1, 0), WD - 1);
            const int cy0 = min(max(y0, 0), HD - 1), cy1 = min(max(y1, 0), HD - 1);
            const int i00 = cy0 * WD + cx0, i01 = cy0 * WD + cx1;
            const int i10 = cy1 * WD + cx0, i11 = cy1 * WD + cx1;
            const float* __restrict__ hb = hm + (size_t)vb * NJ * HW;
#pragma unroll
            for (int j = 0; j < NJ; ++j) {
                const float* __restrict__ hj = hb + j * HW;
                num[j] += w00 * hj[i00] + w01 * hj[i01] + w10 * hj[i10] + w11 * hj[i11];
            }
        }
    }

    // ---- combine views, write cubes (coalesced per joint) ----
    const float inv = fast_rcp(den + 1e-6f);
#pragma unroll
    for (int j = 0; j < NJ; ++j) {
        const float val = fminf(fmaxf(num[j] * inv, 0.0f), 1.0f);
        out[(size_t)(b * NJ + j) * NPTS + n] = val;
    }

    // ---- write grids (broadcast of grid per batch) ----
    float* __restrict__ gout = out + CUBES_ELEMS + ((size_t)b * NPTS + n) * 3;
    gout[0] = gx;
    gout[1] = gy;
    gout[2] = gz;
}

extern "C" void kernel_launch(void* const* d_in, const int* in_sizes, int n_in,
                              void* d_out, int out_size, void* d_ws, size_t ws_size,
                              hipStream_t stream) {
    (void)in_sizes; (void)n_in; (void)out_size; (void)d_ws; (void)ws_size;
    const float* hm   = (const float*)d_in[0];
    const float* camR = (const float*)d_in[1];
    const float* camT = (const float*)d_in[2];
    const float* camF = (const float*)d_in[3];
    const float* camC = (const float*)d_in[4];
    const float* camK = (const float*)d_in[5];
    const float* camP = (const float*)d_in[6];
    const float* ctr  = (const float*)d_in[7];
    const float* scl  = (const float*)d_in[8];
    float* out = (float*)d_out;

    dim3 grid(NPTS / 256, NB);
    project_layer_kernel<<<grid, 256, 0, stream>>>(hm, camR, camT, camF, camC,
                                                   camK, camP, ctr, scl, out);
}